// MoELayer_12610023981276
// MI455X (gfx1250) — compile-verified
//
#include <hip/hip_runtime.h>

#define T_TOKENS 8192
#define DIM      1024
#define HIDDEN   2730
#define NEXP     8
#define LSTRIDE  132   // padded LDS row stride (floats): 528B, 16B-aligned, bank-spread

typedef float v2f __attribute__((ext_vector_type(2)));
typedef float v8f __attribute__((ext_vector_type(8)));

#if defined(__gfx1250__) && __has_builtin(__builtin_amdgcn_global_load_async_to_lds_b128) && __has_builtin(__builtin_amdgcn_s_wait_asynccnt)
#define USE_ASYNC_LDS 1
typedef int v4i_vs __attribute__((vector_size(4 * sizeof(int))));
typedef __attribute__((address_space(1))) v4i_vs* gptr_b128;
typedef __attribute__((address_space(3))) v4i_vs* lptr_b128;
#else
#define USE_ASYNC_LDS 0
#endif

__device__ __forceinline__ v8f wmma4(v2f a, v2f b, v8f c) {
  // D = A(16x4,f32) * B(4x16,f32) + C(16x16,f32)
  return __builtin_amdgcn_wmma_f32_16x16x4_f32(false, a, false, b, (short)0, c,
                                               false, false);
}

__device__ __forceinline__ v8f zero8() {
  v8f z;
#pragma unroll
  for (int i = 0; i < 8; ++i) z[i] = 0.f;
  return z;
}

// ---------------- Gating: one wave32 per token ----------------
__global__ void moe_gate(const float* __restrict__ x, const float* __restrict__ gw,
                         int* __restrict__ tidx, float* __restrict__ tw) {
  int gid  = blockIdx.x * blockDim.x + threadIdx.x;
  int t    = gid >> 5;
  int lane = gid & 31;
  if (t >= T_TOKENS) return;

  float acc[NEXP];
#pragma unroll
  for (int e = 0; e < NEXP; ++e) acc[e] = 0.f;

  const float* xr = x + (size_t)t * DIM;
  for (int d = lane; d < DIM; d += 32) {
    float xv = xr[d];
    const float4* g = (const float4*)(gw + (size_t)d * NEXP);
    float4 g0 = g[0], g1 = g[1];
    acc[0] += xv * g0.x; acc[1] += xv * g0.y;
    acc[2] += xv * g0.z; acc[3] += xv * g0.w;
    acc[4] += xv * g1.x; acc[5] += xv * g1.y;
    acc[6] += xv * g1.z; acc[7] += xv * g1.w;
  }
#pragma unroll
  for (int e = 0; e < NEXP; ++e) {
    float v = acc[e];
    v += __shfl_xor(v, 16, 32);
    v += __shfl_xor(v, 8, 32);
    v += __shfl_xor(v, 4, 32);
    v += __shfl_xor(v, 2, 32);
    v += __shfl_xor(v, 1, 32);
    acc[e] = v;
  }
  if (lane == 0) {
    int e1 = 0;
#pragma unroll
    for (int e = 1; e < NEXP; ++e) if (acc[e] > acc[e1]) e1 = e;
    int e2 = (e1 == 0) ? 1 : 0;
#pragma unroll
    for (int e = 0; e < NEXP; ++e) {
      if (e == e1) continue;
      if (acc[e] > acc[e2]) e2 = e;
    }
    float ex = __expf(acc[e2] - acc[e1]);   // <= 1
    float s  = 1.f + ex;
    tidx[t * 2 + 0] = e1; tidx[t * 2 + 1] = e2;
    tw[t * 2 + 0] = 1.f / s; tw[t * 2 + 1] = ex / s;
  }
}

// ---------------- Scatter tokens into per-expert lists ----------------
__global__ void moe_scatter(const int* __restrict__ tidx, const float* __restrict__ tw,
                            int* __restrict__ cnt, int* __restrict__ list,
                            float* __restrict__ wl) {
  int t = blockIdx.x * blockDim.x + threadIdx.x;
  if (t >= T_TOKENS) return;
#pragma unroll
  for (int k = 0; k < 2; ++k) {
    int e   = tidx[t * 2 + k];
    float w = tw[t * 2 + k];
    int p = atomicAdd(&cnt[e], 1);
    list[(size_t)e * T_TOKENS + p] = t;
    wl  [(size_t)e * T_TOKENS + p] = w;
  }
}

// ---------------- Fused expert FFN: silu(X@w1)*(X@w3) @ w2 ----------------
__global__ __launch_bounds__(256) void moe_ffn(
    const float* __restrict__ x,  const float* __restrict__ w1,
    const float* __restrict__ w2, const float* __restrict__ w3,
    const int* __restrict__ cnt,  const int* __restrict__ list,
    const float* __restrict__ wl, float* __restrict__ out) {
  __shared__ float Xs[32][LSTRIDE];
  __shared__ float Hs[32][LSTRIDE];
  __shared__ int   tok[32];
  __shared__ float twt[32];

  const int e  = blockIdx.y;
  const int mb = blockIdx.x;
  const int ne = cnt[e];
  if (mb * 32 >= ne) return;

  const int tid  = threadIdx.x;
  const int wave = tid >> 5;
  const int lane = tid & 31;
  const int lm   = lane & 15;
  const int koff = (lane >> 4) << 1;  // A/B K sub-offset: 0 or 2
  const int mrow = (lane >> 4) << 3;  // C/D row offset: 0 or 8

  if (tid < 32) {
    int i = mb * 32 + tid;
    if (i < ne) {
      tok[tid] = list[(size_t)e * T_TOKENS + i];
      twt[tid] = wl  [(size_t)e * T_TOKENS + i];
    } else {
      tok[tid] = 0;
      twt[tid] = 0.f;   // padding rows contribute exactly 0
    }
  }
  __syncthreads();

  const float* w1e = w1 + (size_t)e * DIM * HIDDEN;
  const float* w3e = w3 + (size_t)e * DIM * HIDDEN;
  const float* w2e = w2 + (size_t)e * HIDDEN * DIM;

  v8f oacc[2][8];
#pragma unroll
  for (int mt = 0; mt < 2; ++mt)
#pragma unroll
    for (int nt = 0; nt < 8; ++nt) oacc[mt][nt] = zero8();

  const int nchunks = (HIDDEN + 127) / 128;  // 22

  for (int hc = 0; hc < nchunks; ++hc) {
    v8f acc1[2], acc3[2];
#pragma unroll
    for (int mt = 0; mt < 2; ++mt) { acc1[mt] = zero8(); acc3[mt] = zero8(); }

    const int  gcol = hc * 128 + wave * 16 + lm;  // this wave's hidden column
    const bool hok  = gcol < HIDDEN;
    // Clamp address instead of predicating the load: garbage values for
    // out-of-range columns only pollute this lane's own h column, which is
    // masked to zero at the Hs store below. Loads stay unconditional.
    const int  ccol = hok ? gcol : 0;
    const float* bp1 = w1e + ccol;
    const float* bp3 = w3e + ccol;

    // ---- GEMM1 over K = DIM in 8 panels of 128 ----
    for (int kp = 0; kp < 8; ++kp) {
      __syncthreads();
      {  // cooperative gathered X panel load: 32 rows x 128 cols
        int r = tid >> 3, sub = tid & 7;
        const float* src = x + (size_t)tok[r] * DIM + kp * 128 + sub * 16;
        float* dst = &Xs[r][sub * 16];
#if USE_ASYNC_LDS
#pragma unroll
        for (int j = 0; j < 4; ++j)
          __builtin_amdgcn_global_load_async_to_lds_b128(
              (gptr_b128)(src + j * 4), (lptr_b128)(dst + j * 4), 0, 0);
        __builtin_amdgcn_s_wait_asynccnt(0);
#else
#pragma unroll
        for (int j = 0; j < 4; ++j)
          ((float4*)dst)[j] = ((const float4*)src)[j];
#endif
      }
      __syncthreads();
#pragma unroll 4
      for (int k = 0; k < 128; k += 4) {
        int krow = kp * 128 + k + koff;
        v2f b1, b3;
        b1.x = bp1[(size_t)krow * HIDDEN];
        b1.y = bp1[(size_t)(krow + 1) * HIDDEN];
        b3.x = bp3[(size_t)krow * HIDDEN];
        b3.y = bp3[(size_t)(krow + 1) * HIDDEN];
#pragma unroll
        for (int mt = 0; mt < 2; ++mt) {
          v2f a = *(const v2f*)&Xs[mt * 16 + lm][k + koff];
          acc1[mt] = wmma4(a, b1, acc1[mt]);
          acc3[mt] = wmma4(a, b3, acc3[mt]);
        }
      }
    }

    // ---- h = silu(a1) * a3, store (zero-padded) into LDS ----
#pragma unroll
    for (int mt = 0; mt < 2; ++mt) {
#pragma unroll
      for (int r = 0; r < 8; ++r) {
        float a = acc1[mt][r];
        float sig = __builtin_amdgcn_rcpf(1.f + __expf(-a));  // v_rcp_f32
        float h = a * sig * acc3[mt][r];
        Hs[mt * 16 + mrow + r][wave * 16 + lm] = hok ? h : 0.f;
      }
    }
    __syncthreads();

    // ---- GEMM2: oacc += Hs(32x128) @ w2[chunk rows, wave's 128 out cols] ----
    // hrow is always even and HIDDEN is even, so (hrow < HIDDEN) covers both
    // rows of the b64-pair. For hrow >= HIDDEN the matching Hs columns are
    // zero, so clamped-address garbage contributes exactly 0.
#pragma unroll 2
    for (int k = 0; k < 128; k += 4) {
      int hrow = hc * 128 + k + koff;
      int crow = (hrow < HIDDEN) ? hrow : 0;
      v2f a0 = *(const v2f*)&Hs[lm][k + koff];
      v2f a1 = *(const v2f*)&Hs[16 + lm][k + koff];
      const float* bp2 = w2e + (size_t)crow * DIM;
#pragma unroll
      for (int nt = 0; nt < 8; ++nt) {
        int ocol = wave * 128 + nt * 16 + lm;
        v2f b;
        b.x = bp2[ocol];
        b.y = bp2[DIM + ocol];
        oacc[0][nt] = wmma4(a0, b, oacc[0][nt]);
        oacc[1][nt] = wmma4(a1, b, oacc[1][nt]);
      }
    }
    __syncthreads();
  }

  // ---- scale by gate weight, accumulate into output ----
#pragma unroll
  for (int mt = 0; mt < 2; ++mt) {
#pragma unroll
    for (int nt = 0; nt < 8; ++nt) {
#pragma unroll
      for (int r = 0; r < 8; ++r) {
        int trow = mt * 16 + mrow + r;
        int col  = wave * 128 + nt * 16 + lm;
        float v  = oacc[mt][nt][r] * twt[trow];
        atomicAdd(&out[(size_t)tok[trow] * DIM + col], v);
      }
    }
  }
}

extern "C" void kernel_launch(void* const* d_in, const int* in_sizes, int n_in,
                              void* d_out, int out_size, void* d_ws, size_t ws_size,
                              hipStream_t stream) {
  const float* x  = (const float*)d_in[0];
  const float* gw = (const float*)d_in[1];
  const float* w1 = (const float*)d_in[2];
  const float* w2 = (const float*)d_in[3];
  const float* w3 = (const float*)d_in[4];
  float* out = (float*)d_out;

  char* ws = (char*)d_ws;
  int*   tidx = (int*)(ws);            // 2*T ints   = 64 KB
  float* tw   = (float*)(ws + 65536);  // 2*T floats = 64 KB
  int*   cnt  = (int*)(ws + 131072);   // 8 ints (256 B slot)
  int*   list = (int*)(ws + 131328);   // 8*T ints = 256 KB
  float* wl   = (float*)(ws + 393472); // 8*T floats = 256 KB

  (void)hipMemsetAsync(out, 0, (size_t)out_size * sizeof(float), stream);
  (void)hipMemsetAsync(cnt, 0, 256, stream);

  moe_gate<<<T_TOKENS / 8, 256, 0, stream>>>(x, gw, tidx, tw);
  moe_scatter<<<T_TOKENS / 256, 256, 0, stream>>>(tidx, tw, cnt, list, wl);

  dim3 grid(T_TOKENS / 32, NEXP);  // worst-case skew: all tokens on one expert
  moe_ffn<<<grid, 256, 0, stream>>>(x, w1, w2, w3, cnt, list, wl, out);
}